// GAT_55611236549460
// MI455X (gfx1250) — compile-verified
//
#include <hip/hip_runtime.h>
#include <hip/hip_bf16.h>
#include <cstdint>

typedef float v2f __attribute__((ext_vector_type(2)));
typedef float v8f __attribute__((ext_vector_type(8)));
typedef unsigned int u32x4 __attribute__((ext_vector_type(4)));
typedef int i32x4 __attribute__((ext_vector_type(4)));
typedef int i32x8 __attribute__((ext_vector_type(8)));

#define NEG_SLOPE 0.2f

#if defined(__HIP_DEVICE_COMPILE__) && defined(__gfx1250__) && \
    __has_builtin(__builtin_amdgcn_tensor_load_to_lds) &&      \
    __has_builtin(__builtin_amdgcn_s_wait_tensorcnt)
#define USE_TDM 1
#else
#define USE_TDM 0
#endif

__device__ __forceinline__ void atomic_max_float(float* addr, float val) {
  // sign-aware punning: works for mixed-sign values with init = -inf (0xFF800000)
  if (val >= 0.0f) {
    atomicMax((int*)addr, __float_as_int(val));
  } else {
    atomicMin((unsigned int*)addr, __float_as_uint(val));
  }
}

// ---------------- fill ----------------
__global__ void fill_f32(float* __restrict__ p, unsigned int n, float v) {
  unsigned int i = blockIdx.x * blockDim.x + threadIdx.x;
  if (i < n) p[i] = v;
}

// ---------------- GEMM via V_WMMA_F32_16X16X4_F32, weights staged to LDS by TDM ----
// One wave32 per 16x16 output tile of C = A(MxK) * B(KxN), all row-major f32.
// The whole B (weight) matrix (K*Nc floats, <=64KB) is DMA'd into LDS once per
// workgroup with TENSOR_LOAD_TO_LDS; WMMA B operands then come from ds_load.
// A layout: lanes 0-15 -> M=lane, VGPR0/1 = K0,K1 ; lanes 16-31 -> K2,K3.
// C layout: VGPR v, lanes 0-15 -> M=v, N=lane ; lanes 16-31 -> M=v+8.
__global__ void gemm_wmma_f32(const float* __restrict__ A, const float* __restrict__ B,
                              float* __restrict__ C, int M, int K, int Nc) {
  __shared__ float ldsW[128 * 128];  // 64 KB; layer2 uses first 32 KB

#if USE_TDM
  if (threadIdx.x < 32) {  // wave 0 issues the DMA (EXEC ignored for tensor ops)
    unsigned long long ga = (unsigned long long)(uintptr_t)B;
    unsigned lds_base = (unsigned)(uintptr_t)&ldsW[0];  // flat LDS addr low bits = offset
    unsigned td0 = (unsigned)Nc;      // tensor_dim0 (elements)
    unsigned td1 = (unsigned)K;       // tensor_dim1
    unsigned stride0 = (unsigned)Nc;  // tensor_dim0_stride (elements)

    u32x4 g0;
    g0[0] = 1u;                                   // count=1, user descriptor
    g0[1] = lds_base;                             // lds_addr [63:32]
    g0[2] = (unsigned)(ga & 0xFFFFFFFFu);         // global_addr[31:0]
    g0[3] = (unsigned)((ga >> 32) & 0x01FFFFFFu)  // global_addr[56:32]
            | (2u << 30);                         // type=2 ("image")

    i32x8 g1;
    g1[0] = (int)(2u << 16);                      // workgroup_mask=0, data_size=2 (4B)
    g1[1] = (int)((td0 & 0xFFFFu) << 16);         // tensor_dim0[15:0] at [63:48]
    g1[2] = (int)(((td0 >> 16) & 0xFFFFu) | ((td1 & 0xFFFFu) << 16));
    g1[3] = (int)(((td1 >> 16) & 0xFFFFu) | ((td0 & 0xFFFFu) << 16));  // tile_dim0=Nc
    g1[4] = (int)(td1 & 0xFFFFu);                 // tile_dim1=K rows, tile_dim2=0
    g1[5] = (int)stride0;                         // tensor_dim0_stride[31:0]
    g1[6] = 0;                                    // stride0[47:32]=0, dim1_stride unused (2D)
    g1[7] = 0;

    i32x4 gz = {0, 0, 0, 0};
#if __clang_major__ >= 23
    i32x8 gz8 = {0, 0, 0, 0, 0, 0, 0, 0};
    __builtin_amdgcn_tensor_load_to_lds(g0, g1, gz, gz, gz8, 0);
#else
    __builtin_amdgcn_tensor_load_to_lds(g0, g1, gz, gz, 0);
#endif
    __builtin_amdgcn_s_wait_tensorcnt((short)0);
  }
#else
  for (int i = threadIdx.x; i < K * Nc; i += blockDim.x) ldsW[i] = B[i];
#endif
  __syncthreads();

  int wave = blockIdx.x * (blockDim.x >> 5) + (threadIdx.x >> 5);
  int lane = threadIdx.x & 31;
  int tilesN = Nc >> 4;
  int tm = wave / tilesN;
  int tn = wave - tm * tilesN;
  if (tm * 16 >= M) return;  // after barrier: safe
  int hi  = lane >> 4;       // half-wave select (K pairs)
  int l15 = lane & 15;
  const float* arow = A + (size_t)(tm * 16 + l15) * K;
  const float* bcol = &ldsW[tn * 16 + l15];
  v8f acc = {0.f, 0.f, 0.f, 0.f, 0.f, 0.f, 0.f, 0.f};
  for (int k0 = 0; k0 < K; k0 += 4) {
    int ka = k0 + (hi << 1);
    v2f a, b;
    a.x = arow[ka];
    a.y = arow[ka + 1];
    b.x = bcol[ka * Nc];
    b.y = bcol[(ka + 1) * Nc];
    acc = __builtin_amdgcn_wmma_f32_16x16x4_f32(false, a, false, b, (short)0, acc,
                                                false, false);
  }
  float* crow = C + (size_t)(tm * 16 + (hi << 3)) * Nc + tn * 16 + l15;
#pragma unroll
  for (int v = 0; v < 8; ++v) {
    crow[(size_t)v * Nc] = acc[v];
  }
}

// ---------------- per-node attention dots ----------------
// One wave32 per node; lanes striped over H*C channels; butterfly reduce per head group.
template <int H, int C>
__global__ void alpha_kernel(const float* __restrict__ hfeat,
                             const float* __restrict__ a_src,
                             const float* __restrict__ a_dst,
                             float* __restrict__ as, float* __restrict__ ad, int N) {
  const int F = H * C;
  const int PER = F / 32;
  int node = blockIdx.x * (blockDim.x >> 5) + (threadIdx.x >> 5);
  int lane = threadIdx.x & 31;
  if (node >= N) return;
  int f0 = lane * PER;
  int hd = f0 / C;
  const float* hp = hfeat + (size_t)node * F + f0;
  float ps = 0.f, pd = 0.f;
#pragma unroll
  for (int i = 0; i < PER; ++i) {
    float hv = hp[i];
    ps += hv * a_src[f0 + i];
    pd += hv * a_dst[f0 + i];
  }
  const int GRP = 32 / H;  // lanes per head
#pragma unroll
  for (int m = GRP / 2; m >= 1; m >>= 1) {
    ps += __shfl_xor(ps, m, 32);
    pd += __shfl_xor(pd, m, 32);
  }
  if ((lane & (GRP - 1)) == 0) {
    as[node * H + hd] = ps;
    ad[node * H + hd] = pd;
  }
}

// ---------------- edge pass A: segment max of leaky_relu scores ----------------
__global__ void edge_max_kernel(const int* __restrict__ ei, int E, int Etot, int H,
                                const float* __restrict__ as,
                                const float* __restrict__ ad,
                                float* __restrict__ emax) {
  int e = blockIdx.x * blockDim.x + threadIdx.x;
  if (e >= Etot) return;
  int s, d;
  if (e < E) { s = ei[e]; d = ei[E + e]; }
  else       { s = d = e - E; }          // self loops
  for (int hh = 0; hh < H; ++hh) {
    float v = as[s * H + hh] + ad[d * H + hh];
    v = v > 0.f ? v : NEG_SLOPE * v;
    atomic_max_float(&emax[d * H + hh], v);
  }
}

// ---------------- edge pass B: fused exp + denom + weighted scatter ----------------
// One wave32 per edge; lanes striped over H*C channels (float4/float2 per lane).
template <int H, int C>
__global__ void edge_aggregate(const int* __restrict__ ei, int E, int Etot,
                               const float* __restrict__ hfeat,
                               const float* __restrict__ as,
                               const float* __restrict__ ad,
                               const float* __restrict__ emax,
                               float* __restrict__ denom, float* __restrict__ accum) {
  const int F = H * C;
  const int PER = F / 32;
  int wave = blockIdx.x * (blockDim.x >> 5) + (threadIdx.x >> 5);
  int lane = threadIdx.x & 31;
  if (wave >= Etot) return;
  int s, d;
  if (wave < E) { s = ei[wave]; d = ei[E + wave]; }
  else          { s = d = wave - E; }
  int f0 = lane * PER;
  int hd = f0 / C;
  float v = as[s * H + hd] + ad[d * H + hd];
  v = v > 0.f ? v : NEG_SLOPE * v;
  float w = __expf(v - emax[d * H + hd]);
  if ((f0 % C) == 0) atomicAdd(&denom[d * H + hd], w);  // once per edge per head
  const float* hs = hfeat + (size_t)s * F + f0;
  float* ac = accum + (size_t)d * F + f0;
  float vals[PER];
#pragma unroll
  for (int i = 0; i < PER; ++i) vals[i] = hs[i];
#pragma unroll
  for (int i = 0; i < PER; ++i) atomicAdd(&ac[i], vals[i] * w);
}

// ---------------- per-node normalize + bias + ELU (in place) ----------------
__global__ void norm_bias_elu(float* __restrict__ accum, const float* __restrict__ denom,
                              const float* __restrict__ bias, int N, int H, int C) {
  int i = blockIdx.x * blockDim.x + threadIdx.x;
  int F = H * C;
  if (i >= N * F) return;
  int n = i / F;
  int f = i - n * F;
  float v = accum[i] / (denom[n * H + f / C] + 1e-16f) + bias[f];
  accum[i] = v > 0.f ? v : (__expf(v) - 1.f);
}

extern "C" void kernel_launch(void* const* d_in, const int* in_sizes, int n_in,
                              void* d_out, int out_size, void* d_ws, size_t ws_size,
                              hipStream_t stream) {
  const float* x    = (const float*)d_in[0];
  const int*   ei   = (const int*)d_in[1];   // edge_index [2,E]
  const float* W1   = (const float*)d_in[2];
  const float* aS1  = (const float*)d_in[3]; // [2,64]
  const float* aD1  = (const float*)d_in[4];
  const float* b1   = (const float*)d_in[5];
  const float* W2   = (const float*)d_in[6];
  const float* aS2  = (const float*)d_in[7]; // [1,64]
  const float* aD2  = (const float*)d_in[8];
  const float* b2   = (const float*)d_in[9];

  const int IN_C = 128;
  int N = in_sizes[0] / IN_C;
  int E = in_sizes[1] / 2;
  int Etot = E + N;

  // workspace layout (floats)
  float* ws = (float*)d_ws;
  size_t NF1 = (size_t)N * 128;
  float* h1     = ws;                        // N*128
  float* accum1 = h1 + NF1;                  // N*128 (becomes layer-2 input in place)
  float* h2     = accum1 + NF1;              // N*64
  float* as1    = h2 + (size_t)N * 64;       // N*2
  float* ad1    = as1 + (size_t)2 * N;       // N*2
  float* emax1  = ad1 + (size_t)2 * N;       // N*2
  float* emax2  = emax1 + (size_t)2 * N;     // N   (contiguous with emax1 for one fill)
  float* denom1 = emax2 + (size_t)N;         // N*2
  float* denom2 = denom1 + (size_t)2 * N;    // N   (contiguous with denom1)
  float* as2    = denom2 + (size_t)N;        // N
  float* ad2    = as2 + (size_t)N;           // N
  float* out    = (float*)d_out;             // N*64

  const int BT = 256;
  auto nb = [](long n, int t) { return (unsigned)((n + t - 1) / t); };

  // ---- init scratch (every call: deterministic) ----
  fill_f32<<<nb(3L * N, BT), BT, 0, stream>>>(emax1, (unsigned)(3 * N), -INFINITY);
  fill_f32<<<nb(3L * N, BT), BT, 0, stream>>>(denom1, (unsigned)(3 * N), 0.f);
  fill_f32<<<nb((long)NF1, BT), BT, 0, stream>>>(accum1, (unsigned)NF1, 0.f);
  fill_f32<<<nb((long)N * 64, BT), BT, 0, stream>>>(out, (unsigned)(N * 64), 0.f);

  // ---- layer 1: h1 = x @ W1  (M=N, K=128, Nc=128) ----
  {
    long tiles = (long)(N / 16) * (128 / 16);
    gemm_wmma_f32<<<nb(tiles * 32, BT), BT, 0, stream>>>(x, W1, h1, N, 128, 128);
  }
  alpha_kernel<2, 64><<<nb((long)N * 32, BT), BT, 0, stream>>>(h1, aS1, aD1, as1, ad1, N);
  edge_max_kernel<<<nb(Etot, BT), BT, 0, stream>>>(ei, E, Etot, 2, as1, ad1, emax1);
  edge_aggregate<2, 64><<<nb((long)Etot * 32, BT), BT, 0, stream>>>(
      ei, E, Etot, h1, as1, ad1, emax1, denom1, accum1);
  norm_bias_elu<<<nb((long)NF1, BT), BT, 0, stream>>>(accum1, denom1, b1, N, 2, 64);

  // ---- layer 2: h2 = accum1 @ W2  (M=N, K=128, Nc=64) ----
  {
    long tiles = (long)(N / 16) * (64 / 16);
    gemm_wmma_f32<<<nb(tiles * 32, BT), BT, 0, stream>>>(accum1, W2, h2, N, 128, 64);
  }
  alpha_kernel<1, 64><<<nb((long)N * 32, BT), BT, 0, stream>>>(h2, aS2, aD2, as2, ad2, N);
  edge_max_kernel<<<nb(Etot, BT), BT, 0, stream>>>(ei, E, Etot, 1, as2, ad2, emax2);
  edge_aggregate<1, 64><<<nb((long)Etot * 32, BT), BT, 0, stream>>>(
      ei, E, Etot, h2, as2, ad2, emax2, denom2, out);
  norm_bias_elu<<<nb((long)N * 64, BT), BT, 0, stream>>>(out, denom2, b2, N, 1, 64);
}